// All_TransFormer_39144331936314
// MI455X (gfx1250) — compile-verified
//
#include <hip/hip_runtime.h>
#include <hip/hip_bf16.h>

// ---------------------------------------------------------------------------
// All heavy contractions use V_WMMA_F32_16X16X4_F32 (exact fp32 MAC, wave32,
// one wave per 16x16 output tile, EXEC all-ones as the ISA requires).
// Workspace requirement: ~350 MB (buffers reused across pipeline stages).
// ---------------------------------------------------------------------------

typedef float v2f __attribute__((ext_vector_type(2)));
typedef float v8f __attribute__((ext_vector_type(8)));

#define N_   8
#define C_   128
#define T_   64
#define V_   204
#define K_   3
#define H_   3
#define QKD_ 32
#define P_   (T_ * V_)          // 13056 = 816 * 16
#define O_QKV 192               // 2*H*QKD
#define O_OUT 128
#define K_OUT 384               // H*C

static __device__ __forceinline__ v8f wmma4(v2f a, v2f b, v8f c) {
  return __builtin_amdgcn_wmma_f32_16x16x4_f32(false, a, false, b, (short)0, c,
                                               false, false);
}

// --- positional encoding: pe[c*T + t] --------------------------------------
__global__ void k_pe(float* __restrict__ pe) {
  int i = blockIdx.x * blockDim.x + threadIdx.x;
  if (i >= C_ * T_) return;
  int c = i / T_, t = i % T_;
  float div = __expf((float)(2 * (c / 2)) * (-9.210340371976184f / (float)C_));
  float ang = (float)t * div;
  pe[i] = (c & 1) ? __cosf(ang) : __sinf(ang);
}

// --- inverse L2 norm over V per (n,c,t) ------------------------------------
__global__ void k_rnorm(const float* __restrict__ x, float* __restrict__ rn) {
  int i = blockIdx.x * blockDim.x + threadIdx.x;   // (n*C+c)*T + t
  if (i >= N_ * C_ * T_) return;
  const float* row = x + (size_t)i * V_;
  float s = 0.f;
  for (int v = 0; v < V_; ++v) { float a = row[v]; s += a * a; }
  rn[i] = 1.f / fmaxf(sqrtf(s), 1e-12f);
}

// --- dis[n,u,v] = sum_{c,t} xn[.,u]*xn[.,v]  (WMMA, 204 padded to 208) -----
__global__ void k_dis(const float* __restrict__ x, const float* __restrict__ rn,
                      float* __restrict__ dis) {
  int tu = blockIdx.x, tv = blockIdx.y, n = blockIdx.z;
  int lane = threadIdx.x, half = lane >> 4, lm = lane & 15;
  int u = tu * 16 + lm;
  int v = tv * 16 + lm;
  const float* xb = x + (size_t)n * C_ * T_ * V_;   // flat (c*T+t, V)
  const float* rb = rn + n * C_ * T_;
  v8f acc = {};
  for (int k = 0; k < C_ * T_; k += 4) {
    int kk = k + half * 2;
    float r0 = rb[kk], r1 = rb[kk + 1];
    const float* p0 = xb + (size_t)kk * V_;
    v2f a, b;
    a.x = (u < V_) ? p0[u] * r0 : 0.f;
    a.y = (u < V_) ? p0[V_ + u] * r1 : 0.f;
    b.x = (v < V_) ? p0[v] * r0 : 0.f;
    b.y = (v < V_) ? p0[V_ + v] * r1 : 0.f;
    acc = wmma4(a, b, acc);
  }
  for (int j = 0; j < 8; ++j) {
    int m = tu * 16 + j + 8 * half;
    int col = tv * 16 + lm;
    if (m < V_ && col < V_)
      dis[((size_t)n * V_ + m) * V_ + col] = acc[j];
  }
}

// --- top-3 per dis row ------------------------------------------------------
__global__ void k_topk(const float* __restrict__ dis, int* __restrict__ idx) {
  int i = blockIdx.x * blockDim.x + threadIdx.x;   // n*V + u
  if (i >= N_ * V_) return;
  const float* row = dis + (size_t)i * V_;
  float b0 = -1e30f, b1 = -1e30f, b2 = -1e30f;
  int i0 = 0, i1 = 0, i2 = 0;
  for (int v = 0; v < V_; ++v) {
    float d = row[v];
    if (d > b0)      { b2 = b1; i2 = i1; b1 = b0; i1 = i0; b0 = d; i0 = v; }
    else if (d > b1) { b2 = b1; i2 = i1; b1 = d; i1 = v; }
    else if (d > b2) { b2 = d; i2 = v; }
  }
  idx[i * 3 + 0] = i0; idx[i * 3 + 1] = i1; idx[i * 3 + 2] = i2;
}

// --- per (n,v): mean-over-T feats -> mult -> softmax -> w2 -----------------
__global__ void k_mult(const float* __restrict__ x, const int* __restrict__ idx,
                       const float* __restrict__ trans_w,
                       const float* __restrict__ trans_b,
                       const float* __restrict__ convk_w,
                       float* __restrict__ w2) {
  int nv = blockIdx.x;
  int n = nv / V_;
  int c = threadIdx.x;
  __shared__ float m[3][C_];
  __shared__ float mm[3][3];
  int id0 = idx[nv * 3 + 0], id1 = idx[nv * 3 + 1], id2 = idx[nv * 3 + 2];
  const float* base = x + (size_t)(n * C_ + c) * T_ * V_;
  float s0 = 0.f, s1 = 0.f, s2 = 0.f;
  for (int t = 0; t < T_; ++t) {
    const float* r = base + (size_t)t * V_;
    s0 += r[id0]; s1 += r[id1]; s2 += r[id2];
  }
  m[0][c] = s0 * (1.f / T_);
  m[1][c] = s1 * (1.f / T_);
  m[2][c] = s2 * (1.f / T_);
  __syncthreads();
  if (threadIdx.x < 9) {
    int g = threadIdx.x / 3, p = threadIdx.x % 3;
    float s = trans_b[g * 3 + p];
    const float* wr = trans_w + (g * 3 + p) * C_;
    for (int cc = 0; cc < C_; ++cc) s += m[g][cc] * wr[cc];
    mm[g][p] = s;
  }
  __syncthreads();
  if (threadIdx.x < 3) {
    int g = threadIdx.x;
    float a = mm[g][0], b = mm[g][1], d = mm[g][2];
    float mx = fmaxf(a, fmaxf(b, d));
    float e0 = __expf(a - mx), e1 = __expf(b - mx), e2 = __expf(d - mx);
    float inv = 1.f / (e0 + e1 + e2);
    w2[nv * 3 + g] =
        (e0 * convk_w[0] + e1 * convk_w[1] + e2 * convk_w[2]) * inv;
  }
}

// --- gather + pool + ReLU + PE ---------------------------------------------
__global__ void k_pool(const float* __restrict__ x, const int* __restrict__ idx,
                       const float* __restrict__ w2,
                       const float* __restrict__ convk_b,
                       const float* __restrict__ pe,
                       float* __restrict__ xr, float* __restrict__ xs) {
  size_t i = (size_t)blockIdx.x * blockDim.x + threadIdx.x;
  if (i >= (size_t)N_ * C_ * T_ * V_) return;
  int v = (int)(i % V_);
  size_t r = i / V_;            // (n*C+c)*T + t
  int t = (int)(r % T_);
  size_t nc = r / T_;
  int c = (int)(nc % C_);
  int n = (int)(nc / C_);
  const float* row = x + r * V_;
  int base = (n * V_ + v) * 3;
  float s = convk_b[0];
  s += row[idx[base + 0]] * w2[base + 0];
  s += row[idx[base + 1]] * w2[base + 1];
  s += row[idx[base + 2]] * w2[base + 2];
  float xv = fmaxf(s + row[v], 0.f);
  xr[i] = xv;
  xs[i] = xv + pe[c * T_ + t];
}

// --- generic fp32 WMMA GEMM: C[n][m][p] = A[m][k]*B[n][k][p] + bias[m] -----
__global__ void k_gemm(const float* __restrict__ A, const float* __restrict__ B,
                       const float* __restrict__ bias, float* __restrict__ Cm,
                       int M, int Kd) {
  int tp = blockIdx.x, tm = blockIdx.y, n = blockIdx.z;
  int lane = threadIdx.x, half = lane >> 4, lm = lane & 15;
  const float* Bb = B + (size_t)n * Kd * P_;
  float* Cb = Cm + (size_t)n * M * P_;
  int m = tm * 16 + lm;
  int p = tp * 16 + lm;
  const float* Ar = A + (size_t)m * Kd;
  v8f acc = {};
  for (int k = 0; k < Kd; k += 4) {
    int kk = k + half * 2;
    v2f a, b;
    a.x = Ar[kk];
    a.y = Ar[kk + 1];
    b.x = Bb[(size_t)kk * P_ + p];
    b.y = Bb[(size_t)(kk + 1) * P_ + p];
    acc = wmma4(a, b, acc);
  }
  for (int j = 0; j < 8; ++j) {
    int mr = tm * 16 + j + 8 * half;
    Cb[(size_t)mr * P_ + tp * 16 + lm] = acc[j] + bias[mr];
  }
}

// --- att[n,h,t,q] = tanh(q.k/(QKD*V))*alpha + att1s -------------------------
__global__ void k_att(const float* __restrict__ qkv,
                      const float* __restrict__ alphas,
                      const float* __restrict__ att1s,
                      float* __restrict__ att) {
  int tt = blockIdx.x, tq = blockIdx.y, nh = blockIdx.z;
  int n = nh / H_, h = nh % H_;
  int lane = threadIdx.x, half = lane >> 4, lm = lane & 15;
  int trow = tt * 16 + lm;
  int qcol = tq * 16 + lm;
  const float* qb = qkv + ((size_t)n * O_QKV + h * QKD_) * P_;
  const float* kb = qkv + ((size_t)n * O_QKV + H_ * QKD_ + h * QKD_) * P_;
  v8f acc = {};
  for (int k0 = 0; k0 < QKD_ * V_; k0 += 4) {
    int kk = k0 + half * 2;
    int c0 = kk / V_, v0 = kk - c0 * V_;
    int kk1 = kk + 1;
    int c1 = kk1 / V_, v1 = kk1 - c1 * V_;
    v2f a, b;
    a.x = qb[((size_t)c0 * T_ + trow) * V_ + v0];
    a.y = qb[((size_t)c1 * T_ + trow) * V_ + v1];
    b.x = kb[((size_t)c0 * T_ + qcol) * V_ + v0];
    b.y = kb[((size_t)c1 * T_ + qcol) * V_ + v1];
    acc = wmma4(a, b, acc);
  }
  float alpha = alphas[h];
  const float scale = 1.f / (float)(QKD_ * V_);
  for (int j = 0; j < 8; ++j) {
    int m = tt * 16 + j + 8 * half;
    int q = tq * 16 + lm;
    att[(((size_t)n * H_ + h) * T_ + m) * T_ + q] =
        tanhf(acc[j] * scale) * alpha + att1s[((size_t)h * T_ + m) * T_ + q];
  }
}

// --- xs2[n,s,c,q,v] = sum_t x[n,c,t,v]*att[n,s,t,q] (WMMA per (n,s,v)) -----
__global__ void k_xs2(const float* __restrict__ x, const float* __restrict__ att,
                      float* __restrict__ xs2) {
  int tile = blockIdx.x;              // 32 tiles: tq in [0,4), tc in [0,8)
  int tq = tile & 3, tc = tile >> 2;
  int v = blockIdx.y;
  int ns = blockIdx.z;
  int n = ns / H_, s = ns % H_;
  int lane = threadIdx.x, half = lane >> 4, lm = lane & 15;
  int qrow = tq * 16 + lm;
  int ccol = tc * 16 + lm;
  const float* attb = att + ((size_t)n * H_ + s) * T_ * T_;
  const float* xb = x + (size_t)n * C_ * T_ * V_;
  v8f acc = {};
  for (int t0 = 0; t0 < T_; t0 += 4) {
    int t2 = t0 + half * 2;
    v2f a, b;
    a.x = attb[(size_t)t2 * T_ + qrow];
    a.y = attb[(size_t)(t2 + 1) * T_ + qrow];
    b.x = xb[((size_t)ccol * T_ + t2) * V_ + v];
    b.y = xb[((size_t)ccol * T_ + t2 + 1) * V_ + v];
    acc = wmma4(a, b, acc);
  }
  for (int j = 0; j < 8; ++j) {
    int q = tq * 16 + j + 8 * half;
    int c = tc * 16 + lm;
    xs2[((((size_t)n * H_ + s) * C_ + c) * T_ + q) * V_ + v] = acc[j];
  }
}

// --- per-channel BN statistics over (N,T,V) --------------------------------
__global__ void k_bnstats(const float* __restrict__ y, float* __restrict__ mean,
                          float* __restrict__ istd) {
  int c = blockIdx.x;
  __shared__ float s1[256], s2[256];
  float a1 = 0.f, a2 = 0.f;
  for (int i = threadIdx.x; i < N_ * P_; i += blockDim.x) {
    int n = i / P_, p = i - n * P_;
    float val = y[((size_t)n * C_ + c) * P_ + p];
    a1 += val; a2 += val * val;
  }
  s1[threadIdx.x] = a1; s2[threadIdx.x] = a2;
  __syncthreads();
  for (int st = 128; st > 0; st >>= 1) {
    if (threadIdx.x < st) {
      s1[threadIdx.x] += s1[threadIdx.x + st];
      s2[threadIdx.x] += s2[threadIdx.x + st];
    }
    __syncthreads();
  }
  if (threadIdx.x == 0) {
    float inv = 1.f / (float)(N_ * P_);
    float mu = s1[0] * inv;
    float var = s2[0] * inv - mu * mu;
    mean[c] = mu;
    istd[c] = rsqrtf(var + 1e-5f);
  }
}

// --- fused BN + residual + leaky ReLU(0.1) ---------------------------------
__global__ void k_bnact(const float* __restrict__ y,
                        const float* __restrict__ resid,
                        const float* __restrict__ mean,
                        const float* __restrict__ istd,
                        const float* __restrict__ g,
                        const float* __restrict__ beta,
                        float* __restrict__ outp) {
  size_t i = (size_t)blockIdx.x * blockDim.x + threadIdx.x;
  if (i >= (size_t)N_ * C_ * T_ * V_) return;
  int c = (int)((i / P_) % C_);
  float val = (y[i] - mean[c]) * istd[c] * g[c] + beta[c] + resid[i];
  outp[i] = val > 0.f ? val : 0.1f * val;
}

// ---------------------------------------------------------------------------
extern "C" void kernel_launch(void* const* d_in, const int* in_sizes, int n_in,
                              void* d_out, int out_size, void* d_ws,
                              size_t ws_size, hipStream_t stream) {
  const float* x       = (const float*)d_in[0];
  const float* trans_w = (const float*)d_in[1];
  const float* trans_b = (const float*)d_in[2];
  const float* convk_w = (const float*)d_in[3];
  const float* convk_b = (const float*)d_in[4];
  const float* qkv_w   = (const float*)d_in[5];
  const float* qkv_b   = (const float*)d_in[6];
  const float* alphas  = (const float*)d_in[7];
  const float* att1s   = (const float*)d_in[8];
  const float* out_w   = (const float*)d_in[9];
  const float* out_b   = (const float*)d_in[10];
  const float* out_g   = (const float*)d_in[11];
  const float* out_bt  = (const float*)d_in[12];
  const float* ff_w    = (const float*)d_in[13];
  const float* ff_b    = (const float*)d_in[14];
  const float* ff_g    = (const float*)d_in[15];
  const float* ff_bt   = (const float*)d_in[16];
  float* out = (float*)d_out;

  const size_t NCTV = (size_t)N_ * C_ * T_ * V_;        // 13,369,344
  const size_t QKVSZ = (size_t)N_ * O_QKV * P_;         // 20,054,016
  const size_t XS2SZ = (size_t)N_ * K_OUT * P_;         // 40,108,032

  char* ws = (char*)d_ws;
  size_t cur = 0;
  auto alloc = [&](size_t bytes) {
    void* p = ws + cur;
    cur = (cur + bytes + 255) & ~(size_t)255;
    return p;
  };
  float* buf_x   = (float*)alloc(NCTV * 4);   // relu'd x (reused all pipeline)
  float* buf_xs  = (float*)alloc(NCTV * 4);   // x + PE; later reused as ff_raw
  float* buf_qkv = (float*)alloc(QKVSZ * 4);  // qkv; later reused as out_raw
  float* buf_xs2 = (float*)alloc(XS2SZ * 4);  // xs2; later reused as xs3
  float* buf_rn  = (float*)alloc((size_t)N_ * C_ * T_ * 4);
  float* buf_dis = (float*)alloc((size_t)N_ * V_ * V_ * 4);
  int*   buf_idx = (int*)  alloc((size_t)N_ * V_ * 3 * 4);
  float* buf_w2  = (float*)alloc((size_t)N_ * V_ * 3 * 4);
  float* buf_pe  = (float*)alloc((size_t)C_ * T_ * 4);
  float* buf_att = (float*)alloc((size_t)N_ * H_ * T_ * T_ * 4);
  float* buf_mu1 = (float*)alloc(C_ * 4);
  float* buf_is1 = (float*)alloc(C_ * 4);
  float* buf_mu2 = (float*)alloc(C_ * 4);
  float* buf_is2 = (float*)alloc(C_ * 4);
  float* buf_outraw = buf_qkv;                 // qkv dead after k_att
  float* buf_xs3    = buf_xs2;                 // xs2 dead after out GEMM
  float* buf_ffraw  = buf_xs;                  // xs dead after qkv GEMM

  // 1) positional encoding
  k_pe<<<(C_ * T_ + 255) / 256, 256, 0, stream>>>(buf_pe);
  // 2) inverse norms
  k_rnorm<<<(N_ * C_ * T_ + 255) / 256, 256, 0, stream>>>(x, buf_rn);
  // 3) gram matrix (WMMA f32), 204 -> 13 tiles of 16
  k_dis<<<dim3(13, 13, N_), 32, 0, stream>>>(x, buf_rn, buf_dis);
  // 4) top-3 neighbors
  k_topk<<<(N_ * V_ + 255) / 256, 256, 0, stream>>>(buf_dis, buf_idx);
  // 5) transform weights -> softmax -> fold conv-k weights
  k_mult<<<N_ * V_, C_, 0, stream>>>(x, buf_idx, trans_w, trans_b, convk_w,
                                     buf_w2);
  // 6) gather+pool+ReLU (+PE copy for qkv input)
  k_pool<<<(unsigned)((NCTV + 255) / 256), 256, 0, stream>>>(
      x, buf_idx, buf_w2, convk_b, buf_pe, buf_x, buf_xs);
  // 7) qkv projection: (192x128)x(128x13056) per n  (WMMA)
  k_gemm<<<dim3(P_ / 16, O_QKV / 16, N_), 32, 0, stream>>>(
      qkv_w, buf_xs, qkv_b, buf_qkv, O_QKV, C_);
  // 8) attention logits + tanh/alpha/att1s  (WMMA, K = QKD*V = 6528)
  k_att<<<dim3(T_ / 16, T_ / 16, N_ * H_), 32, 0, stream>>>(
      buf_qkv, alphas, att1s, buf_att);
  // 9) apply attention: xs2 (WMMA per (n,s,v))
  k_xs2<<<dim3(32, V_, N_ * H_), 32, 0, stream>>>(buf_x, buf_att, buf_xs2);
  // 10) output projection: (128x384)x(384x13056) per n  (WMMA)
  k_gemm<<<dim3(P_ / 16, O_OUT / 16, N_), 32, 0, stream>>>(
      out_w, buf_xs2, out_b, buf_outraw, O_OUT, K_OUT);
  // 11) BN stats + fused BN/residual/leakyReLU -> xs3
  k_bnstats<<<C_, 256, 0, stream>>>(buf_outraw, buf_mu1, buf_is1);
  k_bnact<<<(unsigned)((NCTV + 255) / 256), 256, 0, stream>>>(
      buf_outraw, buf_x, buf_mu1, buf_is1, out_g, out_bt, buf_xs3);
  // 12) feed-forward: (128x128)x(128x13056) per n  (WMMA)
  k_gemm<<<dim3(P_ / 16, O_OUT / 16, N_), 32, 0, stream>>>(
      ff_w, buf_xs3, ff_b, buf_ffraw, O_OUT, C_);
  // 13) BN stats + fused BN/residual/leakyReLU -> d_out
  k_bnstats<<<C_, 256, 0, stream>>>(buf_ffraw, buf_mu2, buf_is2);
  k_bnact<<<(unsigned)((NCTV + 255) / 256), 256, 0, stream>>>(
      buf_ffraw, buf_x, buf_mu2, buf_is2, ff_g, ff_bt, out);

  (void)in_sizes; (void)n_in; (void)out_size; (void)ws_size;
}